// Attention_Decoder_45586782880447
// MI455X (gfx1250) — compile-verified
//
#include <hip/hip_runtime.h>
#include <hip/hip_bf16.h>

#define H 1024
#define S 4096
#define V 32000
#define NWG 128
#define BLOCK 256
#define WPW (BLOCK / 32)       // waves per workgroup = 8
#define NWAVES (NWG * WPW)     // 1024 waves in grid
#define ROWS_WG (S / NWG)      // 32 encoder rows per WG
#define CH 8                   // keys rows per staged chunk (== WPW)
#define NCH (ROWS_WG / CH)     // 4 chunks per step
#define G_TILES ((3 * H) / 16) // 192 16-row tiles for each GRU matvec
#define OUT_TILES (V / 16)     // 2000 tiles for output projection

typedef __attribute__((ext_vector_type(2))) float v2f;
typedef __attribute__((ext_vector_type(8))) float v8f;

__device__ __forceinline__ float wave_sum(float s) {
#pragma unroll
  for (int off = 16; off > 0; off >>= 1) s += __shfl_xor(s, off, 32);
  return s;
}

// Device-wide sense-reversal barrier. bar[0] = arrive count, bar[16] = generation
// (separate cache lines). All 128 WGs are co-resident.
__device__ __forceinline__ void grid_barrier(unsigned* bar) {
  __threadfence();
  __syncthreads();
  if (threadIdx.x == 0) {
    unsigned gen = __hip_atomic_load(&bar[16], __ATOMIC_RELAXED, __HIP_MEMORY_SCOPE_AGENT);
    unsigned old = __hip_atomic_fetch_add(&bar[0], 1u, __ATOMIC_ACQ_REL, __HIP_MEMORY_SCOPE_AGENT);
    if (old == (unsigned)(NWG - 1)) {
      __hip_atomic_store(&bar[0], 0u, __ATOMIC_RELAXED, __HIP_MEMORY_SCOPE_AGENT);
      __hip_atomic_fetch_add(&bar[16], 1u, __ATOMIC_RELEASE, __HIP_MEMORY_SCOPE_AGENT);
    } else {
      while (__hip_atomic_load(&bar[16], __ATOMIC_ACQUIRE, __HIP_MEMORY_SCOPE_AGENT) == gen)
        __builtin_amdgcn_s_sleep(2);
    }
  }
  __syncthreads();
}

// Issue one 32KB keys chunk (8 rows x 1024 f32, contiguous in global) as
// 8 x global_load_async_to_lds_b128 per thread. Tracked by ASYNCcnt.
// LDS operand is the 32-bit allocation-relative offset (low 32 bits of the
// generic LDS pointer); global address in a VGPR pair, GV mode (saddr = off).
__device__ __forceinline__ void issue_chunk_async(const float* __restrict__ gsrc,
                                                  float* ldst, int tid) {
#pragma unroll
  for (int i = 0; i < (CH * H) / (BLOCK * 4); ++i) {
    int e = (i * BLOCK + tid) * 4;
    unsigned loff = (unsigned)(uintptr_t)(ldst + e);
    unsigned long long gaddr = (unsigned long long)(uintptr_t)(gsrc + e);
    asm volatile("global_load_async_to_lds_b128 %0, %1, off"
                 :: "v"(loff), "v"(gaddr)
                 : "memory");
  }
}

// One wave computes 16 rows of  out[m0..m0+16) = W[m0.., :K] @ x  via
// V_WMMA_F32_16X16X4_F32 with the vector broadcast across all 16 B-columns.
// A layout (32-bit A 16x4): lanes 0-15 hold row M=lane (K0,K1); lanes 16-31 hold
// row M=lane-16 (K2,K3)  -> one float2 load per lane per K-step of 4.
// B mirrors the K striping; broadcast value depends only on K.
// D: VGPR r, lanes 0-15 -> M=r, lanes 16-31 -> M=8+r; lanes 0/16 write out.
__device__ __forceinline__ void wmma_tile_matvec(const float* __restrict__ W, int K,
                                                 const float* xv, int m0,
                                                 float* __restrict__ outp,
                                                 const float* __restrict__ bias) {
  const int l  = threadIdx.x & 31;
  const int lm = l & 15;
  const int kh = l >> 4;  // 0 or 1
  const float* wp = W + (size_t)(m0 + lm) * K + 2 * kh;
  const float* xp = xv + 2 * kh;
  v8f acc = {0.f, 0.f, 0.f, 0.f, 0.f, 0.f, 0.f, 0.f};
  for (int k0 = 0; k0 < K; k0 += 4) {
    v2f a = *(const v2f*)(wp + k0);
    v2f b = *(const v2f*)(xp + k0);
    acc = __builtin_amdgcn_wmma_f32_16x16x4_f32(false, a, false, b, (short)0, acc,
                                                false, false);
  }
  if (lm == 0) {
#pragma unroll
    for (int r = 0; r < 8; ++r) {
      int m = m0 + 8 * kh + r;
      float vv = acc[r];
      if (bias) vv += bias[m];
      outp[m] = vv;
    }
  }
}

// Redundant per-WG GRU elementwise update; bitwise-identical in every WG.
__device__ __forceinline__ void gru_update(const float* __restrict__ gi,
                                           const float* __restrict__ gh,
                                           const float* __restrict__ b_ih,
                                           const float* __restrict__ b_hh,
                                           float* h_lds) {
  const int tid = threadIdx.x;
  float hn[H / BLOCK];
#pragma unroll
  for (int jj = 0; jj < H / BLOCK; ++jj) {
    int i = tid + jj * BLOCK;
    float gir = gi[i] + b_ih[i];
    float giz = gi[i + H] + b_ih[i + H];
    float gin = gi[i + 2 * H] + b_ih[i + 2 * H];
    float ghr = gh[i] + b_hh[i];
    float ghz = gh[i + H] + b_hh[i + H];
    float ghn = gh[i + 2 * H] + b_hh[i + 2 * H];
    float r = 1.f / (1.f + expf(-(gir + ghr)));
    float z = 1.f / (1.f + expf(-(giz + ghz)));
    float n = tanhf(gin + r * ghn);
    hn[jj] = (1.f - z) * n + z * h_lds[i];
  }
  __syncthreads();
#pragma unroll
  for (int jj = 0; jj < H / BLOCK; ++jj) h_lds[tid + jj * BLOCK] = hn[jj];
  __syncthreads();
}

__global__ __launch_bounds__(BLOCK) void attn_gru_decoder_persistent(
    const float* __restrict__ keys, const float* __restrict__ h0,
    const float* __restrict__ W_ih, const float* __restrict__ b_ih,
    const float* __restrict__ W_hh, const float* __restrict__ b_hh,
    const float* __restrict__ W_out, const float* __restrict__ b_out,
    const int* __restrict__ p_out_len, float* __restrict__ out,
    float* __restrict__ partial,   // NWG * H context partials
    float* __restrict__ gi_ws,     // 2 * 3H (double buffered)
    float* __restrict__ gh_ws,     // 2 * 3H (double buffered)
    unsigned* __restrict__ bar) {
  __shared__ __align__(16) float kst[2][CH * H];  // 2 x 32KB async staging
  __shared__ __align__(16) float h_lds[H];
  __shared__ __align__(16) float x_lds[2 * H];
  __shared__ float cos_c[CH];
  __shared__ float knorm_l[ROWS_WG];
  __shared__ float red[WPW];
  __shared__ float hnorm_sh;

  const int tid  = threadIdx.x;
  const int g    = blockIdx.x;
  const int lane = tid & 31;
  const int wv   = tid >> 5;
  const int w    = g * WPW + wv;  // global wave id
  const int srow0 = g * ROWS_WG;
  const int steps = p_out_len[0];

  // ---- init: h0 -> LDS (key norms are fused into step 0's chunk pass)
  for (int j = tid; j < H; j += BLOCK) h_lds[j] = h0[j];
  __syncthreads();

  for (int t = 0; t < steps; ++t) {
    // ---- apply previous step's gates (redundant in every WG)
    if (t > 0)
      gru_update(gi_ws + ((t - 1) & 1) * (3 * H), gh_ws + ((t - 1) & 1) * (3 * H),
                 b_ih, b_hh, h_lds);

    // ---- ||h||
    {
      float s = 0.f;
#pragma unroll
      for (int jj = 0; jj < H / BLOCK; ++jj) {
        float hv = h_lds[tid + jj * BLOCK];
        s += hv * hv;
      }
      s = wave_sum(s);
      if (lane == 0) red[wv] = s;
      __syncthreads();
      if (tid == 0) {
        float tot = 0.f;
        for (int i = 0; i < WPW; ++i) tot += red[i];
        hnorm_sh = sqrtf(tot);
      }
      __syncthreads();
    }

    // ---- attention over our 32 rows: keys staged to LDS ONCE per step via
    // async copies (double-buffered 8-row chunks), consumed by BOTH the cosine
    // dot pass and the context weighted-sum pass.
    float cacc[H / BLOCK] = {0.f, 0.f, 0.f, 0.f};
    issue_chunk_async(keys + (size_t)srow0 * H, kst[0], tid);
    for (int c = 0; c < NCH; ++c) {
      if (c + 1 < NCH)
        issue_chunk_async(keys + (size_t)(srow0 + (c + 1) * CH) * H,
                          kst[(c + 1) & 1], tid);
      if (c + 1 < NCH)
        asm volatile("s_wait_asynccnt 0x8" ::: "memory");  // chunk c landed
      else
        asm volatile("s_wait_asynccnt 0x0" ::: "memory");
      __syncthreads();  // all waves' chunk-c copies resident

      const float* kb = kst[c & 1];
      // cosine weight for row (c*CH + wv): one wave per row, from LDS
      {
        const float* kp = kb + wv * H;
        float s = 0.f, skk = 0.f;
        if (t == 0) {
          for (int j = lane; j < H; j += 32) {
            float kv = kp[j];
            s += kv * h_lds[j];
            skk += kv * kv;
          }
        } else {
          for (int j = lane; j < H; j += 32) s += kp[j] * h_lds[j];
        }
        float kn;
        if (t == 0) {
          skk = wave_sum(skk);
          kn = sqrtf(skk);
          if (lane == 0) knorm_l[c * CH + wv] = kn;
        } else {
          kn = knorm_l[c * CH + wv];
        }
        s = wave_sum(s);
        if (lane == 0) cos_c[wv] = s / (hnorm_sh * kn);
      }
      __syncthreads();
      // context partial accumulate for this chunk (from LDS, deterministic)
#pragma unroll
      for (int jj = 0; jj < H / BLOCK; ++jj) {
        int j = tid + jj * BLOCK;
        float acc = 0.f;
#pragma unroll
        for (int r = 0; r < CH; ++r) acc += cos_c[r] * kb[r * H + j];
        cacc[jj] += acc;
      }
      __syncthreads();  // chunk fully consumed before its buffer is reused
    }
#pragma unroll
    for (int jj = 0; jj < H / BLOCK; ++jj)
      partial[(size_t)g * H + tid + jj * BLOCK] = cacc[jj];

    // ---- W_hh @ h (WMMA tiles spread over all 1024 waves)
    {
      float* gh = gh_ws + (t & 1) * (3 * H);
      int t0 = (w * G_TILES) / NWAVES, t1 = ((w + 1) * G_TILES) / NWAVES;
      for (int tt = t0; tt < t1; ++tt)
        wmma_tile_matvec(W_hh, H, h_lds, tt * 16, gh, nullptr);
    }

    grid_barrier(bar);

    // ---- reduce context partials (fixed order => deterministic), build x
#pragma unroll
    for (int jj = 0; jj < H / BLOCK; ++jj) {
      int j = tid + jj * BLOCK;
      float acc = 0.f;
      for (int gg = 0; gg < NWG; ++gg) acc += partial[(size_t)gg * H + j];
      x_lds[j] = acc;
      x_lds[H + j] = (t == 0) ? 0.f : h_lds[j];  // dec_in
    }
    __syncthreads();

    // ---- W_ih @ x (WMMA tiles, K = 2H)
    {
      float* gi = gi_ws + (t & 1) * (3 * H);
      int t0 = (w * G_TILES) / NWAVES, t1 = ((w + 1) * G_TILES) / NWAVES;
      for (int tt = t0; tt < t1; ++tt)
        wmma_tile_matvec(W_ih, 2 * H, x_lds, tt * 16, gi, nullptr);
    }

    grid_barrier(bar);
  }

  // ---- final gate application, then output projection W_out @ h + b_out
  if (steps > 0)
    gru_update(gi_ws + ((steps - 1) & 1) * (3 * H),
               gh_ws + ((steps - 1) & 1) * (3 * H), b_ih, b_hh, h_lds);

  {
    int t0 = (w * OUT_TILES) / NWAVES, t1 = ((w + 1) * OUT_TILES) / NWAVES;
    for (int tt = t0; tt < t1; ++tt)
      wmma_tile_matvec(W_out, H, h_lds, tt * 16, out, b_out);
  }
}

extern "C" void kernel_launch(void* const* d_in, const int* in_sizes, int n_in,
                              void* d_out, int out_size, void* d_ws, size_t ws_size,
                              hipStream_t stream) {
  const float* keys  = (const float*)d_in[0];  // (S,1,H)
  const float* h0    = (const float*)d_in[1];  // (1,1,H)
  const float* W_ih  = (const float*)d_in[2];  // (3H,2H)
  const float* b_ih  = (const float*)d_in[3];  // (3H,)
  const float* W_hh  = (const float*)d_in[4];  // (3H,H)
  const float* b_hh  = (const float*)d_in[5];  // (3H,)
  const float* W_out = (const float*)d_in[6];  // (V,H)
  const float* b_out = (const float*)d_in[7];  // (V,)
  const int* out_len = (const int*)d_in[8];    // scalar
  float* out = (float*)d_out;

  // workspace layout (floats)
  float* ws      = (float*)d_ws;
  float* partial = ws;                          // NWG*H      = 131072
  float* gi_ws   = partial + (size_t)NWG * H;   // 2*3H       = 6144
  float* gh_ws   = gi_ws + 2 * 3 * H;           // 2*3H       = 6144
  unsigned* bar  = (unsigned*)(gh_ws + 2 * 3 * H);  // 32 uints (2 cache lines)

  // d_ws is poisoned with 0xAA: clear the barrier words (capture-safe memset).
  hipMemsetAsync((void*)bar, 0, 32 * sizeof(unsigned), stream);

  attn_gru_decoder_persistent<<<NWG, BLOCK, 0, stream>>>(
      keys, h0, W_ih, b_ih, W_hh, b_hh, W_out, b_out, out_len, out,
      partial, gi_ws, gh_ws, bar);
}